// ColorTransformsCIELUV_1297080123784
// MI455X (gfx1250) — compile-verified
//
#include <hip/hip_runtime.h>

// Problem constants (fixed shapes from the reference setup).
#define RXD 64
#define RYD 32
#define RZD 32
static constexpr float UP_W = 0.1978f;
static constexpr float VP_W = 0.4683f;
static constexpr float EPSV = 1e-10f;

static constexpr int N_IMG  = 64;
static constexpr int HW     = 224 * 224;     // 50176 floats per channel plane
static constexpr int HW4    = HW / 4;        // 12544 float4 per plane
static constexpr int TOTAL  = N_IMG * HW4;   // 802816 work items (4 px each)
static constexpr int LUT_SZ = RXD * RYD * RZD * 3;

typedef float v4f __attribute__((ext_vector_type(4)));

// Hardware transcendental pow: v_log_f32 + v_exp_f32 (positive x only; callers guard).
// __builtin_amdgcn_logf is log2, __builtin_amdgcn_exp2f is exp2.
__device__ __forceinline__ float fpow(float x, float e) {
    return __builtin_amdgcn_exp2f(e * __builtin_amdgcn_logf(x));
}
__device__ __forceinline__ float frcp(float x) {
    return __builtin_amdgcn_rcpf(x);
}
__device__ __forceinline__ float clamp01(float v) {
    return fminf(fmaxf(v, 0.0f), 1.0f);
}

// sRGB -> linear: v/12.92 if v < 0.04045 else ((v+0.055)/1.055)^2.4
__device__ __forceinline__ float srgb_lin(float v) {
    float lin = v * (1.0f / 12.92f);
    float nl  = fpow(fmaxf(v + 0.055f, 1e-10f) * (1.0f / 1.055f), 2.4f);
    return (v < 0.04045f) ? lin : nl;
}

// linear -> sRGB: 12.92*v if v < 0.0031308 else 1.055*clip(v,1e-10)^(1/2.4)-0.055, clipped [0,1]
__device__ __forceinline__ float srgb_delin(float v) {
    float lin = 12.92f * v;
    float nl  = fmaf(1.055f, fpow(fmaxf(v, 1e-10f), 1.0f / 2.4f), -0.055f);
    return clamp01((v < 0.0031308f) ? lin : nl);
}

__device__ __forceinline__ void lut_acc(const float* __restrict__ q, float w,
                                        float& a0, float& a1, float& a2) {
    a0 = fmaf(w, q[0], a0);
    a1 = fmaf(w, q[1], a1);
    a2 = fmaf(w, q[2], a2);
}

__device__ __forceinline__ void pixel_op(float ri, float gi, float bi,
                                         const float* __restrict__ lut3,
                                         float& ro, float& go, float& bo) {
    // ---- sRGB -> linear RGB -> XYZ ----
    float r = srgb_lin(ri), g = srgb_lin(gi), b = srgb_lin(bi);
    float x = fmaf(0.4124f, r, fmaf(0.3576f, g, 0.1805f * b));
    float y = fmaf(0.2126f, r, fmaf(0.7152f, g, 0.0722f * b));
    float z = fmaf(0.0193f, r, fmaf(0.1192f, g, 0.9504f * b));

    // ---- XYZ -> normalized LUV ----
    float dinv = frcp(fmaf(15.0f, y, fmaf(3.0f, z, x)) + EPSV);
    float up = 4.0f * x * dinv;
    float vp = 9.0f * y * dinv;
    // (6/29)^3 = 0.008856451679; (29/3)^3 = 903.2962963
    float Llin = 903.2962962963f * y;
    float Lcbr = fmaf(116.0f, fpow(fmaxf(y, EPSV), 1.0f / 3.0f), -16.0f);
    float L    = (y < 0.008856451679036f) ? Llin : Lcbr;
    float L13  = 13.0f * L;
    float c0 = L * 0.01f;
    float c1 = fmaf(L13 * (up - UP_W), 0.005f, 0.5f);  // (u+100)/200
    float c2 = fmaf(L13 * (vp - VP_W), 0.005f, 0.5f);  // (v+100)/200

    // ---- Trilinear LUT (clamped indices, unclamped weights) ----
    float sx = c0 * (float)(RXD - 1);
    float sy = c1 * (float)(RYD - 1);
    float sz = c2 * (float)(RZD - 1);
    float fx = floorf(sx), fy = floorf(sy), fz = floorf(sz);
    float px = sx - fx, py = sy - fy, pz = sz - fz;   // == s % 1
    int ix = (int)fx, iy = (int)fy, iz = (int)fz;
    int ix0 = min(max(ix,     0), RXD - 1), ix1 = min(max(ix + 1, 0), RXD - 1);
    int iy0 = min(max(iy,     0), RYD - 1), iy1 = min(max(iy + 1, 0), RYD - 1);
    int iz0 = min(max(iz,     0), RZD - 1), iz1 = min(max(iz + 1, 0), RZD - 1);
    float wx0 = 1.0f - px, wx1 = px;
    float wy0 = 1.0f - py, wy1 = py;
    float wz0 = 1.0f - pz, wz1 = pz;

    int r00 = (ix0 * RYD + iy0) * RZD;
    int r01 = (ix0 * RYD + iy1) * RZD;
    int r10 = (ix1 * RYD + iy0) * RZD;
    int r11 = (ix1 * RYD + iy1) * RZD;

    float a0 = 0.0f, a1 = 0.0f, a2 = 0.0f;
    lut_acc(lut3 + (size_t)(r00 + iz0) * 3, wx0 * wy0 * wz0, a0, a1, a2);
    lut_acc(lut3 + (size_t)(r00 + iz1) * 3, wx0 * wy0 * wz1, a0, a1, a2);
    lut_acc(lut3 + (size_t)(r01 + iz0) * 3, wx0 * wy1 * wz0, a0, a1, a2);
    lut_acc(lut3 + (size_t)(r01 + iz1) * 3, wx0 * wy1 * wz1, a0, a1, a2);
    lut_acc(lut3 + (size_t)(r10 + iz0) * 3, wx1 * wy0 * wz0, a0, a1, a2);
    lut_acc(lut3 + (size_t)(r10 + iz1) * 3, wx1 * wy0 * wz1, a0, a1, a2);
    lut_acc(lut3 + (size_t)(r11 + iz0) * 3, wx1 * wy1 * wz0, a0, a1, a2);
    lut_acc(lut3 + (size_t)(r11 + iz1) * 3, wx1 * wy1 * wz1, a0, a1, a2);
    float o0 = clamp01(a0), o1 = clamp01(a1), o2 = clamp01(a2);

    // ---- normalized LUV -> XYZ ----
    float Lr = o0 * 100.0f;
    float ur = fmaf(o1, 200.0f, -100.0f);
    float vr = fmaf(o2, 200.0f, -100.0f);
    float il  = frcp(fmaf(13.0f, Lr, EPSV));
    float up2 = fmaf(ur, il, UP_W);
    float vp2 = fmaf(vr, il, VP_W);
    // (3/29)^3 = 0.00110705646
    float t116 = (Lr + 16.0f) * (1.0f / 116.0f);
    float ycub = t116 * t116 * t116;
    float yv   = (Lr <= 8.0f) ? (Lr * 0.00110705645988f) : ycub;
    float idn  = frcp(fmaf(4.0f, vp2, EPSV));
    float xv = yv * 9.0f * up2 * idn;
    float zv = yv * (12.0f - fmaf(3.0f, up2, 20.0f * vp2)) * idn;
    xv = fminf(fmaxf(xv, 0.0f), 1.1f);
    float yc = fminf(fmaxf(yv, 0.0f), 1.1f);
    zv = fminf(fmaxf(zv, 0.0f), 1.1f);

    // ---- XYZ -> sRGB ----
    float rr = fmaf(3.2406f, xv, fmaf(-1.5372f, yc, -0.4986f * zv));
    float gg = fmaf(-0.9689f, xv, fmaf(1.8758f, yc, 0.0415f * zv));
    float bb = fmaf(0.0557f, xv, fmaf(-0.2040f, yc, 1.0570f * zv));
    ro = clamp01(srgb_delin(rr));
    go = clamp01(srgb_delin(gg));
    bo = clamp01(srgb_delin(bb));
}

__global__ __launch_bounds__(256)
void luv_lut_kernel(const float* __restrict__ imgs,
                    const float* __restrict__ lut,
                    float* __restrict__ out) {
    int t = blockIdx.x * 256 + threadIdx.x;
    if (t >= TOTAL) return;
    int n = t / HW4;             // image index
    int p = t - n * HW4;         // float4 index within a channel plane

    const float* ib = imgs + (size_t)n * (3 * HW);
    float*       ob = out  + (size_t)n * (3 * HW);
    const float* lut3 = lut + (size_t)n * LUT_SZ;

    // Streaming, once-touched data: non-temporal b128 loads so the 77 MB of
    // image traffic does not evict the 50 MB of L2-resident LUTs.
    v4f r4 = __builtin_nontemporal_load((const v4f*)(ib)           + p);
    v4f g4 = __builtin_nontemporal_load((const v4f*)(ib + HW)      + p);
    v4f b4 = __builtin_nontemporal_load((const v4f*)(ib + 2 * HW)  + p);

    v4f ro, go, bo;
#pragma unroll
    for (int k = 0; k < 4; ++k) {
        float a, b, c;
        pixel_op(r4[k], g4[k], b4[k], lut3, a, b, c);
        ro[k] = a; go[k] = b; bo[k] = c;
    }

    __builtin_nontemporal_store(ro, (v4f*)(ob)          + p);
    __builtin_nontemporal_store(go, (v4f*)(ob + HW)     + p);
    __builtin_nontemporal_store(bo, (v4f*)(ob + 2 * HW) + p);
}

extern "C" void kernel_launch(void* const* d_in, const int* in_sizes, int n_in,
                              void* d_out, int out_size, void* d_ws, size_t ws_size,
                              hipStream_t stream) {
    const float* imgs = (const float*)d_in[0];   // (64, 3, 224, 224) f32
    const float* lut  = (const float*)d_in[1];   // (64, 64, 32, 32, 3) f32
    float* out = (float*)d_out;                  // (64, 3, 224, 224) f32
    (void)in_sizes; (void)n_in; (void)out_size; (void)d_ws; (void)ws_size;

    const int blocks = (TOTAL + 255) / 256;      // 3136
    luv_lut_kernel<<<blocks, 256, 0, stream>>>(imgs, lut, out);
}